// fpn_classifier_graph_48146583388684
// MI455X (gfx1250) — compile-verified
//
#include <hip/hip_runtime.h>
#include <hip/hip_bf16.h>
#include <math.h>

typedef float v2f __attribute__((ext_vector_type(2)));
typedef float v8f __attribute__((ext_vector_type(8)));

#define N_ROIS     2000
#define M_PAD      2048
#define POOL       7
#define K1         12544      // 256*7*7
#define NF         1024
#define NCLS       81
#define NBOX       324        // 81*4
#define N3         405        // 81 + 324
#define N3_PAD     512
#define BN_EPS     0.001f

// GEMM tiling
#define KC         16         // K-chunk staged through LDS (divides 12544 & 1024)
#define SK         20         // padded LDS row stride (floats): 16B aligned, 16 distinct banks
#define ATILE      (128 * SK) // 2560 floats per operand tile
#define LDSBUF     (2 * ATILE)// A tile + B tile per buffer

// ---------------- workspace layout (floats) ----------------
#define WS_POOLED  ((size_t)0)
#define WS_H1      (WS_POOLED + (size_t)M_PAD * K1)
#define WS_SHARED  (WS_H1     + (size_t)M_PAD * NF)
#define WS_W3      (WS_SHARED + (size_t)M_PAD * NF)
#define WS_B3      (WS_W3     + (size_t)N3_PAD * NF)
#define WS_OUT3    (WS_B3     + (size_t)N3_PAD)

// =====================================================================
// 1) ROI-align: pooled[n][c*49 + i*7 + j], rows >= N_ROIS zero-padded
// grid (M_PAD, 49) x 256 threads
// =====================================================================
__global__ __launch_bounds__(256)
void roi_pool_kernel(const float* __restrict__ p2, const float* __restrict__ p3,
                     const float* __restrict__ p4, const float* __restrict__ p5,
                     const float* __restrict__ rois, float* __restrict__ pooled)
{
    const int n  = blockIdx.x;
    const int k  = blockIdx.y * 256 + threadIdx.x;   // 0..12543
    if (n >= N_ROIS) { pooled[(size_t)n * K1 + k] = 0.f; return; }

    const float x1 = rois[n * 4 + 0];
    const float y1 = rois[n * 4 + 1];
    const float x2 = rois[n * 4 + 2];
    const float y2 = rois[n * 4 + 3];

    const float bw = x2 - x1, bh = y2 - y1;
    float lv = rintf(log2f(sqrtf(bh * bw) / 224.f)) + 4.f;
    lv = fminf(fmaxf(lv, 2.f), 5.f);
    const int lvl = (int)lv;

    const float* fm; int H;
    if      (lvl == 2) { fm = p2; H = 256; }
    else if (lvl == 3) { fm = p3; H = 128; }
    else if (lvl == 4) { fm = p4; H = 64;  }
    else               { fm = p5; H = 32;  }

    const int c  = k / 49;
    const int ij = k - c * 49;
    const int i  = ij / POOL;
    const int j  = ij - i * POOL;

    const float scale = (float)(H - 1) / 1024.f;     // normalize then *(H-1)
    const float ti = (float)i / 6.f, tj = (float)j / 6.f;
    const float ys = y1 * scale + ti * (y2 - y1) * scale;
    const float xs = x1 * scale + tj * (x2 - x1) * scale;

    const float y0f = floorf(ys), x0f = floorf(xs);
    const float ly = ys - y0f, lx = xs - x0f;
    int y0 = (int)y0f; y0 = max(0, min(y0, H - 1));
    int x0 = (int)x0f; x0 = max(0, min(x0, H - 1));
    const int y1i = min(y0 + 1, H - 1);
    const int x1i = min(x0 + 1, H - 1);

    const float* base = fm + (size_t)c * H * H;
    const float v00 = base[y0  * H + x0 ];
    const float v01 = base[y0  * H + x1i];
    const float v10 = base[y1i * H + x0 ];
    const float v11 = base[y1i * H + x1i];

    const float val = v00 * (1.f - ly) * (1.f - lx)
                    + v01 * (1.f - ly) * lx
                    + v10 * ly * (1.f - lx)
                    + v11 * ly * lx;
    pooled[(size_t)n * K1 + k] = val;
}

// =====================================================================
// 2) pack logits_w/bbox_w (+biases) into padded W3 [512 x 1024], b3[512]
// =====================================================================
__global__ __launch_bounds__(256)
void pack_w3_kernel(const float* __restrict__ lw, const float* __restrict__ lb,
                    const float* __restrict__ bw, const float* __restrict__ bb,
                    float* __restrict__ w3, float* __restrict__ b3)
{
    const int idx = blockIdx.x * 256 + threadIdx.x;  // 0..512*1024-1
    const int row = idx >> 10;
    const int col = idx & 1023;
    float v = 0.f;
    if (row < NCLS)      v = lw[row * NF + col];
    else if (row < N3)   v = bw[(row - NCLS) * NF + col];
    w3[idx] = v;
    if (col == 0) {
        float b = 0.f;
        if (row < NCLS)    b = lb[row];
        else if (row < N3) b = bb[row - NCLS];
        b3[row] = b;
    }
}

// =====================================================================
// 3) WMMA fp32 GEMM with LDS double buffering:
//    C[M x N] = A[M x K] * B[N x K]^T  (+ epilogue)
//    EPI==0: y = relu((x + bias - bm) * g * rsqrt(bv+eps) + bbta)
//    EPI==1: y = x + bias
// Block tile 128x128 (8 waves as 4M x 2N, wave tile 32x64). Each KC=16
// chunk of A and B is staged once per block into LDS (40 KB, double
// buffered); next chunk's global b128 loads are issued before this
// chunk's 32 v_wmma ops, ds_store'd after, one barrier per chunk.
// Fragment reads are bank-conflict-free ds_load_b64 (stride 20).
// Requirements: M,N multiples of 128; K multiple of 16.
// =====================================================================
template <int EPI>
__global__ __launch_bounds__(256)
void gemm_wmma_f32(const float* __restrict__ A, const float* __restrict__ B,
                   float* __restrict__ C, int N, int K,
                   const float* __restrict__ bias, const float* __restrict__ g,
                   const float* __restrict__ bbta, const float* __restrict__ bm,
                   const float* __restrict__ bv)
{
    __shared__ float smem[2 * LDSBUF];   // 40 KB

    const int tid  = threadIdx.x;
    const int wave = tid >> 5;
    const int lane = tid & 31;
    const int wm   = wave & 3;          // wave row (0..3)
    const int wn   = wave >> 2;         // wave col (0..1)
    const int m0b  = blockIdx.x * 128;
    const int n0b  = blockIdx.y * 128;
    const int half = lane >> 4;         // selects K pair {0,1} vs {2,3}
    const int r    = lane & 15;

    // ---- staging addresses: 256 threads cover 128 rows x 16 cols, twice ----
    const int trow = tid >> 2;          // 0..63
    const int tcol = (tid & 3) << 2;    // 0,4,8,12
    const float* gA0 = A + (size_t)(m0b + trow) * K + tcol;
    const float* gA1 = gA0 + (size_t)64 * K;
    const float* gB0 = B + (size_t)(n0b + trow) * K + tcol;
    const float* gB1 = gB0 + (size_t)64 * K;
    const int s0 = trow * SK + tcol;         // LDS offset within a tile
    const int s1 = (trow + 64) * SK + tcol;

    // ---- fragment LDS offsets (within A/B tile) ----
    const int aOff = (wm * 32 + r) * SK + 2 * half;
    const int bOff = (wn * 64 + r) * SK + 2 * half;

    v8f acc[2][4];
#pragma unroll
    for (int mi = 0; mi < 2; ++mi)
#pragma unroll
        for (int nj = 0; nj < 4; ++nj)
#pragma unroll
            for (int e = 0; e < 8; ++e) acc[mi][nj][e] = 0.f;

    auto loadF = [&](const float* sa, const float* sb, int kk,
                     v2f af[2], v2f bf[4]) {
        af[0] = *(const v2f*)(sa + kk);
        af[1] = *(const v2f*)(sa + 16 * SK + kk);
        bf[0] = *(const v2f*)(sb + kk);
        bf[1] = *(const v2f*)(sb + 16 * SK + kk);
        bf[2] = *(const v2f*)(sb + 32 * SK + kk);
        bf[3] = *(const v2f*)(sb + 48 * SK + kk);
    };
    auto mma = [&](v2f af[2], v2f bf[4]) {
#pragma unroll
        for (int mi = 0; mi < 2; ++mi)
#pragma unroll
            for (int nj = 0; nj < 4; ++nj)
                acc[mi][nj] = __builtin_amdgcn_wmma_f32_16x16x4_f32(
                    false, af[mi], false, bf[nj], (short)0, acc[mi][nj],
                    false, false);
    };

    // ---- preload chunk 0 into buffer 0 ----
    {
        float4 ra0 = *(const float4*)gA0;
        float4 ra1 = *(const float4*)gA1;
        float4 rb0 = *(const float4*)gB0;
        float4 rb1 = *(const float4*)gB1;
        *(float4*)(smem + s0)         = ra0;
        *(float4*)(smem + s1)         = ra1;
        *(float4*)(smem + ATILE + s0) = rb0;
        *(float4*)(smem + ATILE + s1) = rb1;
    }
    __syncthreads();

    int buf = 0;
    for (int kc = 0; kc < K; kc += KC) {
        const bool more = (kc + KC) < K;
        float4 ra0, ra1, rb0, rb1;
        if (more) {                       // issue next chunk's global loads now
            ra0 = *(const float4*)(gA0 + kc + KC);
            ra1 = *(const float4*)(gA1 + kc + KC);
            rb0 = *(const float4*)(gB0 + kc + KC);
            rb1 = *(const float4*)(gB1 + kc + KC);
        }

        // ---- compute this chunk from LDS (register double-buffered) ----
        const float* sa = smem + buf * LDSBUF + aOff;
        const float* sb = smem + buf * LDSBUF + ATILE + bOff;
        v2f aE[2], bE[4], aO[2], bO[4];
        loadF(sa, sb, 0,  aE, bE);
        loadF(sa, sb, 4,  aO, bO); mma(aE, bE);
        loadF(sa, sb, 8,  aE, bE); mma(aO, bO);
        loadF(sa, sb, 12, aO, bO); mma(aE, bE);
        mma(aO, bO);

        if (more) {                       // store staged chunk into other buffer
            float* d = smem + (buf ^ 1) * LDSBUF;
            *(float4*)(d + s0)         = ra0;
            *(float4*)(d + s1)         = ra1;
            *(float4*)(d + ATILE + s0) = rb0;
            *(float4*)(d + ATILE + s1) = rb1;
        }
        __syncthreads();
        buf ^= 1;
    }

    // epilogue + store. C/D layout: VGPR v, lanes 0-15 -> M=v, lanes 16-31 -> M=v+8
#pragma unroll
    for (int nj = 0; nj < 4; ++nj) {
        const int col = n0b + wn * 64 + nj * 16 + r;
        float sc, sh;
        if (EPI == 0) {
            sc = g[col] * rsqrtf(bv[col] + BN_EPS);
            sh = (bias[col] - bm[col]) * sc + bbta[col];
        } else {
            sc = 1.f; sh = bias[col];
        }
#pragma unroll
        for (int mi = 0; mi < 2; ++mi) {
            const size_t rowbase =
                (size_t)(m0b + wm * 32 + mi * 16 + half * 8) * N + col;
#pragma unroll
            for (int v = 0; v < 8; ++v) {
                float x = acc[mi][nj][v] * sc + sh;
                if (EPI == 0) x = fmaxf(x, 0.f);
                C[rowbase + (size_t)v * N] = x;
            }
        }
    }
}

// =====================================================================
// 4) softmax over 81 logits per row + scatter logits/probs/bbox into d_out
// one block (128 threads) per ROI row
// =====================================================================
__global__ __launch_bounds__(128)
void softmax_scatter_kernel(const float* __restrict__ out3, float* __restrict__ dout)
{
    __shared__ float red[128];
    const int n   = blockIdx.x;
    const int tid = threadIdx.x;
    const float* row = out3 + (size_t)n * N3_PAD;

    float v = (tid < NCLS) ? row[tid] : -INFINITY;
    red[tid] = v; __syncthreads();
#pragma unroll
    for (int s = 64; s > 0; s >>= 1) {
        if (tid < s) red[tid] = fmaxf(red[tid], red[tid + s]);
        __syncthreads();
    }
    const float mx = red[0]; __syncthreads();

    const float e = (tid < NCLS) ? expf(v - mx) : 0.f;
    red[tid] = e; __syncthreads();
#pragma unroll
    for (int s = 64; s > 0; s >>= 1) {
        if (tid < s) red[tid] += red[tid + s];
        __syncthreads();
    }
    const float inv = 1.f / red[0];

    if (tid < NCLS) {
        dout[(size_t)n * NCLS + tid] = v;                                  // logits
        dout[(size_t)N_ROIS * NCLS + (size_t)n * NCLS + tid] = e * inv;    // probs
    }
    const size_t bbox_base = (size_t)2 * N_ROIS * NCLS;
    for (int idx = tid; idx < NBOX; idx += 128)
        dout[bbox_base + (size_t)n * NBOX + idx] = row[NCLS + idx];
}

// =====================================================================
extern "C" void kernel_launch(void* const* d_in, const int* in_sizes, int n_in,
                              void* d_out, int out_size, void* d_ws, size_t ws_size,
                              hipStream_t stream)
{
    const float* p2      = (const float*)d_in[0];
    const float* p3      = (const float*)d_in[1];
    const float* p4      = (const float*)d_in[2];
    const float* p5      = (const float*)d_in[3];
    const float* rois    = (const float*)d_in[4];
    const float* conv1_w = (const float*)d_in[5];
    const float* conv1_b = (const float*)d_in[6];
    const float* bn1_g   = (const float*)d_in[7];
    const float* bn1_b   = (const float*)d_in[8];
    const float* bn1_m   = (const float*)d_in[9];
    const float* bn1_v   = (const float*)d_in[10];
    const float* conv2_w = (const float*)d_in[11];
    const float* conv2_b = (const float*)d_in[12];
    const float* bn2_g   = (const float*)d_in[13];
    const float* bn2_b   = (const float*)d_in[14];
    const float* bn2_m   = (const float*)d_in[15];
    const float* bn2_v   = (const float*)d_in[16];
    const float* lw      = (const float*)d_in[17];
    const float* lb      = (const float*)d_in[18];
    const float* bw      = (const float*)d_in[19];
    const float* bb      = (const float*)d_in[20];

    float* ws     = (float*)d_ws;
    float* pooled = ws + WS_POOLED;
    float* h1     = ws + WS_H1;
    float* shared = ws + WS_SHARED;
    float* w3     = ws + WS_W3;
    float* b3     = ws + WS_B3;
    float* out3   = ws + WS_OUT3;
    float* dout   = (float*)d_out;

    // 1) ROI align -> pooled [2048 x 12544] (rows >= 2000 zeroed)
    roi_pool_kernel<<<dim3(M_PAD, K1 / 256), 256, 0, stream>>>(
        p2, p3, p4, p5, rois, pooled);

    // 2) pack padded head weights
    pack_w3_kernel<<<(N3_PAD * NF) / 256, 256, 0, stream>>>(lw, lb, bw, bb, w3, b3);

    // 3) GEMM1: pooled x conv1_w^T -> h1  (+BN1+ReLU)
    gemm_wmma_f32<0><<<dim3(M_PAD / 128, NF / 128), 256, 0, stream>>>(
        pooled, conv1_w, h1, NF, K1, conv1_b, bn1_g, bn1_b, bn1_m, bn1_v);

    // 4) GEMM2: h1 x conv2_w^T -> shared  (+BN2+ReLU)
    gemm_wmma_f32<0><<<dim3(M_PAD / 128, NF / 128), 256, 0, stream>>>(
        h1, conv2_w, shared, NF, NF, conv2_b, bn2_g, bn2_b, bn2_m, bn2_v);

    // 5) GEMM3: shared x W3^T -> out3 [2048 x 512]  (+bias)
    gemm_wmma_f32<1><<<dim3(M_PAD / 128, N3_PAD / 128), 256, 0, stream>>>(
        shared, w3, out3, N3_PAD, NF, b3, nullptr, nullptr, nullptr, nullptr);

    // 6) softmax + scatter into d_out
    softmax_scatter_kernel<<<N_ROIS, 128, 0, stream>>>(out3, dout);
}